// Fingerprint_38534446579800
// MI455X (gfx1250) — compile-verified
//
#include <hip/hip_runtime.h>

#define D_      128
#define NA_     4096
#define NB_     8192
#define NM_     128
#define ENC_N   16384   // D*D

typedef __attribute__((ext_vector_type(2))) float v2f;
typedef __attribute__((ext_vector_type(8))) float v8f;

#define AS1 __attribute__((address_space(1)))
#define AS3 __attribute__((address_space(3)))

#if defined(__has_builtin)
#if __has_builtin(__builtin_amdgcn_global_load_async_to_lds_b32) && \
    __has_builtin(__builtin_amdgcn_s_wait_asynccnt)
#define USE_ASYNC_LDS 1
#endif
#endif
#ifndef USE_ASYNC_LDS
#define USE_ASYNC_LDS 0
#endif

// ---------------------------------------------------------------------------
// device helpers
// ---------------------------------------------------------------------------
__device__ __forceinline__ float eluf(float x)   { return x > 0.f ? x : expm1f(x); }
__device__ __forceinline__ float sigf(float x)   { return 1.f / (1.f + __expf(-x)); }
__device__ __forceinline__ float lreluf(float x) { return x > 0.f ? x : 0.01f * x; }

__device__ __forceinline__ unsigned fkey(float f) {
    unsigned u = __float_as_uint(f);
    return (u & 0x80000000u) ? ~u : (u | 0x80000000u);
}
__device__ __forceinline__ float fdec(unsigned k) {
    unsigned u = (k & 0x80000000u) ? (k ^ 0x80000000u) : ~k;
    return __uint_as_float(u);
}

// ---------------------------------------------------------------------------
// generic fp32 WMMA GEMM:  Out[M x N] = act( bn( A[M x K] * B + bias ) )
// One wave computes a 16x32 tile (two 16x16 accumulators) so each A fragment
// feeds two V_WMMA_F32_16X16X4_F32 ops. All mode flags are template params so
// the inner loop is pure load + wmma (no branches, no EXEC divergence).
// BTRANS:  B element (k,n) read as B[n*ldb+k]  (GRU W^T GEMMs)
// SCATTER: atomicAdd(scatterOut[seg[row]*ldo+col], aw[row]*val)
// TAIL:    handle K%4 with one branchless masked step (only the K=39 GEMM)
// ---------------------------------------------------------------------------
template <int BTRANS, int SCATTER, int RELU, int TAIL>
__global__ void wmma_gemm_kernel(
    const float* __restrict__ A, int K, int lda,
    const float* __restrict__ B, int ldb,
    const float* __restrict__ bias,
    const float* __restrict__ bnScale, const float* __restrict__ bnShift,
    float* __restrict__ Out, int ldo,
    const float* __restrict__ aw, const int* __restrict__ seg, int segStride,
    float* __restrict__ scatterOut)
{
    const int lane = threadIdx.x & 31;
    const int hh   = lane >> 4;      // lane half (0/1)
    const int l15  = lane & 15;
    const int rowA  = blockIdx.x * 16 + l15;
    const int colB0 = blockIdx.y * 32 + l15;
    const int colB1 = colB0 + 16;

    const float* __restrict__ Ap  = A + (size_t)rowA * lda;
    const float* __restrict__ Bp0 = BTRANS ? (B + (size_t)colB0 * ldb) : (B + colB0);
    const float* __restrict__ Bp1 = BTRANS ? (B + (size_t)colB1 * ldb) : (B + colB1);

    v8f acc0 = {}, acc1 = {};
    const int K4 = TAIL ? (K & ~3) : K;
    for (int k0 = 0; k0 < K4; k0 += 4) {
        const int ka = k0 + hh * 2;
        v2f av, b0, b1;
        av.x = Ap[ka];
        av.y = Ap[ka + 1];
        if (BTRANS) {
            b0.x = Bp0[ka];  b0.y = Bp0[ka + 1];
            b1.x = Bp1[ka];  b1.y = Bp1[ka + 1];
        } else {
            b0.x = Bp0[(size_t)(ka    ) * ldb];
            b0.y = Bp0[(size_t)(ka + 1) * ldb];
            b1.x = Bp1[(size_t)(ka    ) * ldb];
            b1.y = Bp1[(size_t)(ka + 1) * ldb];
        }
        acc0 = __builtin_amdgcn_wmma_f32_16x16x4_f32(false, av, false, b0,
                                                     (short)0, acc0, false, false);
        acc1 = __builtin_amdgcn_wmma_f32_16x16x4_f32(false, av, false, b1,
                                                     (short)0, acc1, false, false);
    }
    if (TAIL && K4 < K) {   // branchless masked tail (K=39 path only)
        const int ka  = K4 + hh * 2;
        const int ka0 = ka     < K ? ka     : K - 1;
        const int ka1 = ka + 1 < K ? ka + 1 : K - 1;
        const float m0 = ka     < K ? 1.f : 0.f;
        const float m1 = ka + 1 < K ? 1.f : 0.f;
        v2f av, b0, b1;
        av.x = Ap[ka0] * m0;
        av.y = Ap[ka1] * m1;
        if (BTRANS) {
            b0.x = Bp0[ka0] * m0;  b0.y = Bp0[ka1] * m1;
            b1.x = Bp1[ka0] * m0;  b1.y = Bp1[ka1] * m1;
        } else {
            b0.x = Bp0[(size_t)ka0 * ldb] * m0;
            b0.y = Bp0[(size_t)ka1 * ldb] * m1;
            b1.x = Bp1[(size_t)ka0 * ldb] * m0;
            b1.y = Bp1[(size_t)ka1 * ldb] * m1;
        }
        acc0 = __builtin_amdgcn_wmma_f32_16x16x4_f32(false, av, false, b0,
                                                     (short)0, acc0, false, false);
        acc1 = __builtin_amdgcn_wmma_f32_16x16x4_f32(false, av, false, b1,
                                                     (short)0, acc1, false, false);
    }

    // epilogue for both 16x16 halves of the 16x32 tile
#pragma unroll
    for (int t = 0; t < 2; ++t) {
        const int colG = blockIdx.y * 32 + t * 16 + l15;
        const v8f& acc = t ? acc1 : acc0;
        float sc = 1.f, sh = 0.f;
        if (bnScale)   { sc = bnScale[colG]; sh = bnShift[colG]; }
        else if (bias) { sh = bias[colG]; }
#pragma unroll
        for (int i = 0; i < 8; ++i) {
            const int rowG = blockIdx.x * 16 + i + 8 * hh;
            float v = fmaf(acc[i], sc, sh);
            if (RELU) v = fmaxf(v, 0.f);
            if (SCATTER) {
                const int s = seg[rowG * segStride];
                atomicAdd(&scatterOut[(size_t)s * ldo + colG], aw[rowG] * v);
            } else {
                Out[(size_t)rowG * ldo + colG] = v;
            }
        }
    }
}

// ---------------------------------------------------------------------------
// fused per-edge kernel:  neighbor[e,f] = sum_d x[dst[e],d] *
//                         relu( bn( bond[e,:] @ Wenc[:, d*128+f] ) )
// 16 edges / block, 8 waves; wave w owns f-tile [w*16, w*16+16).
// Bm (8192x128x128) is never materialized. All pad-guards are branchless.
// The 16 gathered x[dst] rows are staged to LDS via the async-to-LDS unit
// when the toolchain exposes it (no VGPR round-trip), else via plain stores.
// ---------------------------------------------------------------------------
__global__ void __launch_bounds__(256)
neighbor_kernel(const float* __restrict__ x,
                const float* __restrict__ bond,
                const int*   __restrict__ bidx,
                const float* __restrict__ Wenc,
                const float* __restrict__ esc,   // enc bn scale  (16384)
                const float* __restrict__ esh,   // enc bn shift  (16384, bias folded)
                float* __restrict__ neighbor)
{
    __shared__ float sA[16 * D_];                // dst atom rows (8 KB)
    const int eBase = blockIdx.x * 16;
    const int tid   = threadIdx.x;

#if USE_ASYNC_LDS
    for (int t = tid; t < 16 * D_; t += 256) {
        const int m    = t >> 7;
        const int dcol = t & 127;
        const int dst  = bidx[2 * (eBase + m) + 1];
        const float* g = x + (size_t)dst * D_ + dcol;
        AS1 int* gp = (AS1 int*)(unsigned long long)g;                 // global ptr
        AS3 int* lp = (AS3 int*)(unsigned)(unsigned long long)&sA[t];  // LDS offset
        __builtin_amdgcn_global_load_async_to_lds_b32(gp, lp, 0, 0);
    }
    __builtin_amdgcn_s_wait_asynccnt(0);
    __syncthreads();
#else
    for (int t = tid; t < 16 * D_; t += 256) {
        const int m    = t >> 7;
        const int dcol = t & 127;
        const int dst  = bidx[2 * (eBase + m) + 1];
        sA[t] = x[(size_t)dst * D_ + dcol];
    }
    __syncthreads();
#endif

    const int lane = tid & 31;
    const int wave = tid >> 5;
    const int hh   = lane >> 4;
    const int l15  = lane & 15;
    const int f0   = wave * 16;

    // A fragments: bond rows (16 x 10, K padded to 12), branchless, loop-invariant
    v2f af[3];
    const int erow = eBase + l15;
#pragma unroll
    for (int s = 0; s < 3; ++s) {
        const int k0  = s * 4 + hh * 2;             // even, in {0,2,4,6,8,10}
        const int r0  = k0 < 10 ? k0 : 8;           // clamp into valid rows
        const float mk = k0 < 10 ? 1.f : 0.f;       // k0<10 implies k0+1<10 here
        af[s].x = bond[erow * 10 + r0    ] * mk;
        af[s].y = bond[erow * 10 + r0 + 1] * mk;
    }

    v8f accN = {};
    for (int d = 0; d < D_; ++d) {
        const int col = d * D_ + f0 + l15;
        v8f t8 = {};
#pragma unroll
        for (int st = 0; st < 3; ++st) {
            const int k0  = st * 4 + hh * 2;
            const int r0  = k0 < 10 ? k0 : 8;
            const float mk = k0 < 10 ? 1.f : 0.f;
            v2f bv;
            bv.x = Wenc[(size_t)(r0    ) * ENC_N + col] * mk;
            bv.y = Wenc[(size_t)(r0 + 1) * ENC_N + col] * mk;
            t8 = __builtin_amdgcn_wmma_f32_16x16x4_f32(false, af[st], false, bv,
                                                       (short)0, t8, false, false);
        }
        const float sc = esc[col];
        const float sh = esh[col];
#pragma unroll
        for (int i = 0; i < 8; ++i) {
            float tv = fmaxf(fmaf(t8[i], sc, sh), 0.f);      // relu(bn(enc))
            const float av = sA[(i + 8 * hh) * D_ + d];      // x[dst][d] (LDS broadcast)
            accN[i] = fmaf(tv, av, accN[i]);
        }
    }
#pragma unroll
    for (int i = 0; i < 8; ++i)
        neighbor[(size_t)(eBase + i + 8 * hh) * D_ + f0 + l15] = accN[i];
}

// ---------------------------------------------------------------------------
// small elementwise / reduction kernels
// ---------------------------------------------------------------------------
__global__ void zero_kernel(unsigned* p, int n) {
    int i = blockIdx.x * blockDim.x + threadIdx.x;
    if (i < n) p[i] = 0u;
}

__global__ void elu_kernel(const float* __restrict__ in, float* __restrict__ out, int n) {
    int i = blockIdx.x * blockDim.x + threadIdx.x;
    if (i < n) out[i] = eluf(in[i]);
}

__global__ void bnprep_kernel(const float* b, const float* g, const float* be,
                              const float* m, const float* v,
                              float* sc, float* sh, int n) {
    int i = blockIdx.x * blockDim.x + threadIdx.x;
    if (i < n) {
        float s = g[i] * rsqrtf(v[i] + 1e-6f);
        sc[i] = s;
        sh[i] = (b[i] - m[i]) * s + be[i];
    }
}

__global__ void __launch_bounds__(256)
prop_score_kernel(const float* __restrict__ x, const float* __restrict__ nb,
                  const int* __restrict__ bidx, const float* __restrict__ aW,
                  const float* __restrict__ ab, float* __restrict__ score, int n) {
    const int e    = blockIdx.x * 8 + (threadIdx.x >> 5);
    const int lane = threadIdx.x & 31;
    if (e >= n) return;
    const int src = bidx[2 * e];
    float s = 0.f;
    for (int j = lane; j < D_; j += 32)
        s += x[(size_t)src * D_ + j] * aW[j] + nb[(size_t)e * D_ + j] * aW[D_ + j];
    for (int o = 16; o; o >>= 1) s += __shfl_xor(s, o, 32);
    if (lane == 0) score[e] = lreluf(s + ab[0]);
}

__global__ void __launch_bounds__(256)
super_score_kernel(const float* __restrict__ sup, const float* __restrict__ x,
                   const int* __restrict__ mol, const float* __restrict__ aW,
                   const float* __restrict__ ab, float* __restrict__ score, int n) {
    const int a    = blockIdx.x * 8 + (threadIdx.x >> 5);
    const int lane = threadIdx.x & 31;
    if (a >= n) return;
    const int m = mol[a];
    float s = 0.f;
    for (int j = lane; j < D_; j += 32)
        s += sup[(size_t)m * D_ + j] * aW[j] + x[(size_t)a * D_ + j] * aW[D_ + j];
    for (int o = 16; o; o >>= 1) s += __shfl_xor(s, o, 32);
    if (lane == 0) score[a] = lreluf(s + ab[0]);
}

__global__ void segmax_kernel(const float* s, const int* idx, int stride,
                              unsigned* mx, int n) {
    int i = blockIdx.x * blockDim.x + threadIdx.x;
    if (i < n) atomicMax(&mx[idx[i * stride]], fkey(s[i]));
}

__global__ void expsum_kernel(const float* s, const int* idx, int stride,
                              const unsigned* mx, float* w, float* sum, int n) {
    int i = blockIdx.x * blockDim.x + threadIdx.x;
    if (i < n) {
        const int g = idx[i * stride];
        const float e = __expf(s[i] - fdec(mx[g]));
        w[i] = e;
        atomicAdd(&sum[g], e);
    }
}

__global__ void aw_kernel(const float* w, const int* idx, int stride,
                          const float* sum, float* aw, int n) {
    int i = blockIdx.x * blockDim.x + threadIdx.x;
    if (i < n) aw[i] = w[i] / (sum[idx[i * stride]] + 1e-8f);
}

__global__ void scatter_rows_kernel(const float* __restrict__ x,
                                    const int* __restrict__ mol,
                                    float* __restrict__ out, int rows) {
    int i = blockIdx.x * blockDim.x + threadIdx.x;
    if (i < rows * D_) {
        const int a = i >> 7, j = i & 127;
        atomicAdd(&out[(size_t)mol[a] * D_ + j], x[i]);
    }
}

__global__ void gru_kernel(const float* __restrict__ gi, const float* __restrict__ gh,
                           const float* __restrict__ h, float* __restrict__ out, int rows) {
    int i = blockIdx.x * blockDim.x + threadIdx.x;
    if (i >= rows * D_) return;
    const int a = i >> 7, j = i & 127;
    const float* gia = gi + (size_t)a * 3 * D_;
    const float* gha = gh + (size_t)a * 3 * D_;
    const float r  = sigf(gia[j]           + gha[j]);
    const float z  = sigf(gia[D_ + j]      + gha[D_ + j]);
    const float nn = tanhf(gia[2 * D_ + j] + r * gha[2 * D_ + j]);
    out[i] = (1.f - z) * nn + z * h[i];
}

__global__ void __launch_bounds__(256)
pred_kernel(const float* __restrict__ h, const float* __restrict__ pw,
            const float* __restrict__ pb, float* __restrict__ out, int rows) {
    const int r    = blockIdx.x * 8 + (threadIdx.x >> 5);
    const int lane = threadIdx.x & 31;
    if (r >= rows) return;
    float s = 0.f;
    for (int j = lane; j < D_; j += 32) s += h[(size_t)r * D_ + j] * pw[j];
    for (int o = 16; o; o >>= 1) s += __shfl_xor(s, o, 32);
    if (lane == 0) out[r] = s + pb[0];
}

// ---------------------------------------------------------------------------
// host side
// ---------------------------------------------------------------------------
struct LBN  { const float *W, *b, *be, *g, *m, *v; };
struct GRUP { const float *Wih, *Whh, *bih, *bhh; };
struct PropP  { const float *aW, *ab; LBN att, enc; GRUP gru; };
struct SuperP { const float *aW, *ab; LBN att; GRUP gru; };

static inline void zero_n(void* p, int n, hipStream_t st) {
    zero_kernel<<<(n + 255) / 256, 256, 0, st>>>((unsigned*)p, n);
}

static inline void bnprep(const LBN& p, float* sc, float* sh, int n, hipStream_t st) {
    bnprep_kernel<<<(n + 255) / 256, 256, 0, st>>>(p.b, p.g, p.be, p.m, p.v, sc, sh, n);
}

// store GEMMs -------------------------------------------------------------
static inline void gemm_pre(hipStream_t st, const float* A, int M, int K, int lda,
                            const float* B, int ldb, const float* sc, const float* sh,
                            float* Out, int N, int ldo) {
    dim3 g(M / 16, N / 32);
    wmma_gemm_kernel<0, 0, 1, 1><<<g, 32, 0, st>>>(A, K, lda, B, ldb, nullptr,
                                                   sc, sh, Out, ldo,
                                                   nullptr, nullptr, 0, nullptr);
}
static inline void gemm_bnrelu(hipStream_t st, const float* A, int M, int K, int lda,
                               const float* B, int ldb, const float* sc, const float* sh,
                               float* Out, int N, int ldo) {
    dim3 g(M / 16, N / 32);
    wmma_gemm_kernel<0, 0, 1, 0><<<g, 32, 0, st>>>(A, K, lda, B, ldb, nullptr,
                                                   sc, sh, Out, ldo,
                                                   nullptr, nullptr, 0, nullptr);
}
static inline void gemm_bias_T(hipStream_t st, const float* A, int M, int K, int lda,
                               const float* B, int ldb, const float* bias,
                               float* Out, int N, int ldo) {
    dim3 g(M / 16, N / 32);
    wmma_gemm_kernel<1, 0, 0, 0><<<g, 32, 0, st>>>(A, K, lda, B, ldb, bias,
                                                   nullptr, nullptr, Out, ldo,
                                                   nullptr, nullptr, 0, nullptr);
}
static inline void gemm_scatter(hipStream_t st, const float* A, int M, int K, int lda,
                                const float* B, int ldb, const float* sc, const float* sh,
                                int N, int ldo, const float* aw, const int* seg,
                                int segStride, float* scat) {
    dim3 g(M / 16, N / 32);
    wmma_gemm_kernel<0, 1, 0, 0><<<g, 32, 0, st>>>(A, K, lda, B, ldb, nullptr,
                                                   sc, sh, nullptr, ldo,
                                                   aw, seg, segStride, scat);
}

extern "C" void kernel_launch(void* const* d_in, const int* in_sizes, int n_in,
                              void* d_out, int out_size, void* d_ws, size_t ws_size,
                              hipStream_t stream) {
    auto F = [&](int i) { return (const float*)d_in[i]; };
    auto I = [&](int i) { return (const int*)d_in[i]; };

    // Detect flattening order of the nested input dict:
    //   sorted (JAX pytree) : in_sizes[2] == NB*2   (bond_index)
    //   insertion (dict)    : in_sizes[2] == 39*128 (pre.W)
    const bool sorted = (n_in > 2) && (in_sizes[2] == NB_ * 2);

    const float *atom, *bond;
    const int *bidx, *mol;
    LBN preL, pred1L;
    const float *predW, *predb;
    PropP prop[3];
    SuperP sup[3];

    if (sorted) {
        atom = F(0); bond = F(1); bidx = I(2); mol = I(3);
        preL   = { F(4),  F(5),  F(6),  F(7),  F(8),  F(9)  };
        pred1L = { F(10), F(11), F(12), F(13), F(14), F(15) };
        predW = F(16); predb = F(17);
        for (int k = 0; k < 3; ++k) {
            int b0 = 18 + 18 * k;
            prop[k].aW = F(b0 + 0); prop[k].ab = F(b0 + 1);
            prop[k].att = { F(b0+2), F(b0+3), F(b0+4), F(b0+5), F(b0+6), F(b0+7) };
            prop[k].enc = { F(b0+8), F(b0+9), F(b0+10), F(b0+11), F(b0+12), F(b0+13) };
            prop[k].gru = { F(b0+15), F(b0+14), F(b0+17), F(b0+16) }; // list order: Whh,Wih,bhh,bih
        }
        for (int t = 0; t < 3; ++t) {
            int b0 = 72 + 12 * t;
            sup[t].aW = F(b0 + 0); sup[t].ab = F(b0 + 1);
            sup[t].att = { F(b0+2), F(b0+3), F(b0+4), F(b0+5), F(b0+6), F(b0+7) };
            sup[t].gru = { F(b0+9), F(b0+8), F(b0+11), F(b0+10) };
        }
    } else {
        atom = F(0); bond = F(1);
        preL = { F(2), F(3), F(5), F(4), F(6), F(7) };      // stored order: W,b,g,be,m,v
        for (int k = 0; k < 3; ++k) {
            int b0 = 8 + 18 * k;
            prop[k].enc = { F(b0+0), F(b0+1), F(b0+3), F(b0+2), F(b0+4), F(b0+5) };
            prop[k].aW = F(b0 + 6); prop[k].ab = F(b0 + 7);
            prop[k].att = { F(b0+8), F(b0+9), F(b0+11), F(b0+10), F(b0+12), F(b0+13) };
            prop[k].gru = { F(b0+14), F(b0+15), F(b0+16), F(b0+17) };
        }
        for (int t = 0; t < 3; ++t) {
            int b0 = 62 + 12 * t;
            sup[t].aW = F(b0 + 0); sup[t].ab = F(b0 + 1);
            sup[t].att = { F(b0+2), F(b0+3), F(b0+5), F(b0+4), F(b0+6), F(b0+7) };
            sup[t].gru = { F(b0+8), F(b0+9), F(b0+10), F(b0+11) };
        }
        pred1L = { F(98), F(99), F(101), F(100), F(102), F(103) };
        predW = F(104); predb = F(105);
        bidx = I(106); mol = I(107);
    }
    (void)ws_size;

    // workspace carve-up (floats)
    float* ws = (float*)d_ws;
    size_t o = 0;
    auto alloc = [&](size_t n) { float* p = ws + o; o += n; return p; };
    float* x0     = alloc((size_t)NA_ * D_);
    float* x1     = alloc((size_t)NA_ * D_);
    float* nb     = alloc((size_t)NB_ * D_);
    float* ctx    = alloc((size_t)NA_ * D_);
    float* ctxe   = alloc((size_t)NA_ * D_);     // elu(ctx)
    float* gi     = alloc((size_t)NA_ * 3 * D_);
    float* gh     = alloc((size_t)NA_ * 3 * D_);
    float* sup0   = alloc((size_t)NM_ * D_);
    float* sup1   = alloc((size_t)NM_ * D_);
    float* hpred  = alloc((size_t)NM_ * D_);
    float* score  = alloc(NB_);
    float* wexp   = alloc(NB_);
    float* awb    = alloc(NB_);
    unsigned* smx = (unsigned*)alloc(NA_);
    float* ssum   = alloc(NA_);
    float* encsc  = alloc(ENC_N);
    float* encsh  = alloc(ENC_N);
    float* smsc   = alloc(D_);
    float* smsh   = alloc(D_);

    // --- pre: x = relu(bn(atom @ Wpre + b)) ---
    bnprep(preL, smsc, smsh, D_, stream);
    gemm_pre(stream, atom, NA_, 39, 39, preL.W, D_, smsc, smsh, x0, D_, D_);

    float* x = x0; float* xn = x1;

    // --- K graph-attention layers ---
    for (int k = 0; k < 3; ++k) {
        const PropP& P = prop[k];
        bnprep(P.enc, encsc, encsh, ENC_N, stream);
        neighbor_kernel<<<NB_ / 16, 256, 0, stream>>>(x, bond, bidx, P.enc.W,
                                                      encsc, encsh, nb);
        prop_score_kernel<<<NB_ / 8, 256, 0, stream>>>(x, nb, bidx, P.aW, P.ab, score, NB_);
        zero_n(smx, NA_, stream);
        zero_n(ssum, NA_, stream);
        segmax_kernel<<<NB_ / 256, 256, 0, stream>>>(score, bidx, 2, smx, NB_);
        expsum_kernel<<<NB_ / 256, 256, 0, stream>>>(score, bidx, 2, smx, wexp, ssum, NB_);
        aw_kernel<<<NB_ / 256, 256, 0, stream>>>(wexp, bidx, 2, ssum, awb, NB_);
        bnprep(P.att, smsc, smsh, D_, stream);
        zero_n(ctx, NA_ * D_, stream);
        gemm_scatter(stream, nb, NB_, D_, D_, P.att.W, D_, smsc, smsh,
                     D_, D_, awb, bidx, 2, ctx);                        // scatter into ctx
        elu_kernel<<<(NA_ * D_) / 256, 256, 0, stream>>>(ctx, ctxe, NA_ * D_);
        gemm_bias_T(stream, ctxe, NA_, D_, D_, P.gru.Wih, D_, P.gru.bih,
                    gi, 3 * D_, 3 * D_);                                // gi = elu(ctx)@Wih^T
        gemm_bias_T(stream, x,    NA_, D_, D_, P.gru.Whh, D_, P.gru.bhh,
                    gh, 3 * D_, 3 * D_);                                // gh = x@Whh^T
        gru_kernel<<<(NA_ * D_) / 256, 256, 0, stream>>>(gi, gh, x, xn, NA_);
        float* t = x; x = xn; xn = t;
    }

    // --- superatom init: segment_sum(x, mol) ---
    zero_n(sup0, NM_ * D_, stream);
    scatter_rows_kernel<<<(NA_ * D_) / 256, 256, 0, stream>>>(x, mol, sup0, NA_);

    float* sp = sup0; float* spn = sup1;

    // --- T super-attention layers ---
    for (int t = 0; t < 3; ++t) {
        const SuperP& S = sup[t];
        super_score_kernel<<<NA_ / 8, 256, 0, stream>>>(sp, x, mol, S.aW, S.ab, score, NA_);
        zero_n(smx, NM_, stream);
        zero_n(ssum, NM_, stream);
        segmax_kernel<<<NA_ / 256, 256, 0, stream>>>(score, mol, 1, smx, NA_);
        expsum_kernel<<<NA_ / 256, 256, 0, stream>>>(score, mol, 1, smx, wexp, ssum, NA_);
        aw_kernel<<<NA_ / 256, 256, 0, stream>>>(wexp, mol, 1, ssum, awb, NA_);
        bnprep(S.att, smsc, smsh, D_, stream);
        zero_n(ctx, NM_ * D_, stream);
        gemm_scatter(stream, x, NA_, D_, D_, S.att.W, D_, smsc, smsh,
                     D_, D_, awb, mol, 1, ctx);                         // scatter into ctx_s
        elu_kernel<<<(NM_ * D_) / 256, 256, 0, stream>>>(ctx, ctxe, NM_ * D_);
        gemm_bias_T(stream, ctxe, NM_, D_, D_, S.gru.Wih, D_, S.gru.bih,
                    gi, 3 * D_, 3 * D_);
        gemm_bias_T(stream, sp,   NM_, D_, D_, S.gru.Whh, D_, S.gru.bhh,
                    gh, 3 * D_, 3 * D_);
        gru_kernel<<<(NM_ * D_) / 256, 256, 0, stream>>>(gi, gh, sp, spn, NM_);
        float* tt = sp; sp = spn; spn = tt;
    }

    // --- head: out = relu(bn(sup @ W1)) @ predW + predb ---
    bnprep(pred1L, smsc, smsh, D_, stream);
    gemm_bnrelu(stream, sp, NM_, D_, D_, pred1L.W, D_, smsc, smsh, hpred, D_, D_);
    pred_kernel<<<NM_ / 8, 256, 0, stream>>>(hpred, predW, predb, (float*)d_out, NM_);
    (void)out_size;
}